// FeedForwardQuantum_65481071405464
// MI455X (gfx1250) — compile-verified
//
#include <hip/hip_runtime.h>

#define N_QUBITS 8
#define DIM 256       // 2^8
#define EMBED 512
#define FFN 2048

typedef __attribute__((ext_vector_type(2))) float v2f;
typedef __attribute__((ext_vector_type(8))) float v8f;

// ---------------------------------------------------------------------------
// Kernel 1: compute z[8] from U[:,0], then h[f] = relu(b1[f] + sum_q z_q W1[f,q])
// One block, 256 threads (8 waves). Output h -> ws[0..2047].
// ---------------------------------------------------------------------------
__global__ __launch_bounds__(256) void quantum_h_kernel(
    const float* __restrict__ U_re, const float* __restrict__ U_im,
    const float* __restrict__ W1,   const float* __restrict__ b1,
    float* __restrict__ h_out)
{
    __shared__ float s_probs[DIM];
    __shared__ float s_z[N_QUBITS];
    const int tid = threadIdx.x;

    // psi = U[:,0]; probs_k = re^2 + im^2 (column 0, row stride DIM)
    {
        float re = U_re[(size_t)tid * DIM];
        float im = U_im[(size_t)tid * DIM];
        s_probs[tid] = re * re + im * im;
    }
    __syncthreads();

    // z_i = sum_k probs[k] * (bit_{7-i}(k) ? -1 : +1)
    if (tid < N_QUBITS) {
        const int shift = (N_QUBITS - 1) - tid;
        float z = 0.0f;
        for (int k = 0; k < DIM; ++k) {
            float p = s_probs[k];
            z += ((k >> shift) & 1) ? -p : p;
        }
        s_z[tid] = z;
    }
    __syncthreads();

    // h[f] = relu(b1[f] + z . W1[f,:])
    float z[N_QUBITS];
#pragma unroll
    for (int q = 0; q < N_QUBITS; ++q) z[q] = s_z[q];

    for (int f = tid; f < FFN; f += 256) {
        const float* w = W1 + (size_t)f * N_QUBITS;
        float acc = b1[f];
#pragma unroll
        for (int q = 0; q < N_QUBITS; ++q) acc = fmaf(z[q], w[q], acc);
        h_out[f] = fmaxf(acc, 0.0f);
    }
}

// ---------------------------------------------------------------------------
// Kernel 2: vec[e] = b2[e] + sum_f h[f] * W2[e,f] via V_WMMA_F32_16X16X4_F32.
// A(16x4) = h-slice replicated in all 16 rows; B(4x16) = 16 rows of W2.
// Grid = EMBED/16 = 32 blocks; 8 waves per block split K=2048 into 256-chunks
// (64 chained WMMAs each), then LDS reduction of accumulator row M=0.
// ---------------------------------------------------------------------------
__global__ __launch_bounds__(256) void gemv_wmma_kernel(
    const float* __restrict__ h_in, const float* __restrict__ W2,
    const float* __restrict__ b2,   float* __restrict__ vec)
{
    __shared__ float sh[FFN];
    __shared__ float s_part[8][16];

    const int tid  = threadIdx.x;
    const int wave = tid >> 5;
    const int lane = tid & 31;
    const int n    = lane & 15;        // N column within tile
    const int khalf = (lane >> 4) * 2; // lanes 0-15: K=0,1 ; lanes 16-31: K=2,3
    const int e0   = blockIdx.x * 16;

    // stage h into LDS (broadcast reuse by all 8 waves)
    for (int f = tid; f < FFN; f += 256) sh[f] = h_in[f];
    __syncthreads();

    const float* w2row = W2 + (size_t)(e0 + n) * FFN;
    const int kstart = wave * (FFN / 8); // 256 K-values per wave

    v8f c = {};
    for (int k0 = 0; k0 < FFN / 8; k0 += 4) {
        const int k = kstart + k0 + khalf;
        // A: row of h (identical for all M rows) -> lane holds A[M][khalf..khalf+1]
        v2f a = *(const v2f*)(sh + k);
        // B: B[K][N=n] = W2[e0+n, k] -> lane holds B[khalf..khalf+1][n]
        v2f b = *(const v2f*)(w2row + k);
        c = __builtin_amdgcn_wmma_f32_16x16x4_f32(
                false, a, false, b, (short)0, c, false, false);
    }

    // D row M=0 lives in accumulator VGPR0 on lanes 0..15 (N = lane)
    if (lane < 16) s_part[wave][n] = c[0];
    __syncthreads();

    if (tid < 16) {
        float acc = b2[e0 + tid];
#pragma unroll
        for (int w = 0; w < 8; ++w) acc += s_part[w][tid];
        vec[e0 + tid] = acc;
    }
}

// ---------------------------------------------------------------------------
// Kernel 3: broadcast vec[512] to out (B*S copies). Pure store bandwidth.
// blockDim=256 -> each thread's float4 target index mod 128 is fixed, so the
// value is loaded ONCE into registers; the loop is stores only (b128).
// ---------------------------------------------------------------------------
__global__ __launch_bounds__(256) void broadcast_kernel(
    const float* __restrict__ vec, float4* __restrict__ out, int n4)
{
    const int tid = threadIdx.x;
    // float4 index within the repeating 512-float (128-float4) pattern
    const float4 v = reinterpret_cast<const float4*>(vec)[tid & 127];

    int idx = blockIdx.x * blockDim.x + tid;
    const int stride = gridDim.x * blockDim.x; // *4 floats -> multiple of 512
    for (; idx < n4; idx += stride) {
        out[idx] = v;
    }
}

// ---------------------------------------------------------------------------
extern "C" void kernel_launch(void* const* d_in, const int* in_sizes, int n_in,
                              void* d_out, int out_size, void* d_ws, size_t ws_size,
                              hipStream_t stream)
{
    // inputs: 0:x (unused!), 1:U_re, 2:U_im, 3:W1, 4:b1, 5:W2, 6:b2
    const float* U_re = (const float*)d_in[1];
    const float* U_im = (const float*)d_in[2];
    const float* W1   = (const float*)d_in[3];
    const float* b1   = (const float*)d_in[4];
    const float* W2   = (const float*)d_in[5];
    const float* b2   = (const float*)d_in[6];

    float* ws  = (float*)d_ws;
    float* h   = ws;          // FFN floats
    float* vec = ws + FFN;    // EMBED floats

    quantum_h_kernel<<<1, 256, 0, stream>>>(U_re, U_im, W1, b1, h);
    gemv_wmma_kernel<<<EMBED / 16, 256, 0, stream>>>(h, W2, b2, vec);

    const int n4 = out_size / 4; // float4 count (out_size = 4*8192*512)
    broadcast_kernel<<<4096, 256, 0, stream>>>(vec, (float4*)d_out, n4);
}